// ST_BLOCK_0_15513421873580
// MI455X (gfx1250) — compile-verified
//
#include <hip/hip_runtime.h>
#include <math.h>

// ---------------------------------------------------------------------------
// Problem dimensions (compile-time; match reference)
// ---------------------------------------------------------------------------
#define Bn   32
#define CIN  64
#define COUT 64
#define Nn   307
#define Tn   24
#define Kc   3
#define NT   (Nn * Tn)          // 7368
#define CNT  (COUT * Nn * Tn)   // 471552
#define NNn  (Nn * Nn)          // 94249

typedef __attribute__((ext_vector_type(2))) float v2f;
typedef __attribute__((ext_vector_type(8))) float v8f;

// ---------------------------------------------------------------------------
// Generic fp32 WMMA GEMM: C[z] = A[z] * B[z] (+bias)(+relu)
//   One wave computes a 32x32 tile of C as 2x2 blocks of 16x16 WMMA tiles
//   (V_WMMA_F32_16X16X4_F32, K stepped by 4). Out-of-range rows/cols are
//   handled by CLAMPING addresses (garbage lands in accumulator slots that
//   the epilogue never stores), so the hot loop is branch-free and EXEC
//   stays all-ones as WMMA requires. Only the K%4 remainder step masks A.
//   Batch offset = (z/zdiv)*s_o + (z%zdiv)*s_i per operand so one kernel
//   covers per-b, per-(b,k) and per-(b,n) batched GEMMs.
// ---------------------------------------------------------------------------
__device__ __forceinline__ void store_tile(float* __restrict__ C, int ldc,
                                           const v8f& acc, int mbase, int nbase,
                                           int M, int N, const float* bias,
                                           int relu, int half, int lid) {
    const int col = nbase + lid;
    if (col >= N) return;
    for (int r = 0; r < 8; ++r) {
        const int mr = mbase + r + 8 * half;   // C/D layout: M = r + 8*half
        if (mr < M) {
            float v = acc[r];
            if (bias) v += bias[mr];
            if (relu) v = fmaxf(v, 0.f);
            C[(long)mr * ldc + col] = v;
        }
    }
}

__global__ __launch_bounds__(256)
void wmma_gemm_f32(const float* __restrict__ A,
                   const float* __restrict__ B,
                   float* __restrict__ C,
                   const float* __restrict__ bias,
                   int M, int N, int K,
                   int lda, int ldb, int ldc,
                   long sAo, long sAi, long sBo, long sBi,
                   long sCo, long sCi, int zdiv, int relu)
{
    const int z  = blockIdx.z;
    const long zo = z / zdiv, zi = z % zdiv;
    A += zo * sAo + zi * sAi;
    B += zo * sBo + zi * sBi;
    C += zo * sCo + zi * sCi;

    const int tilesN = (N + 31) >> 5;
    const int tilesM = (M + 31) >> 5;
    const int wave   = (int)((blockIdx.x * blockDim.x + threadIdx.x) >> 5);
    if (wave >= tilesM * tilesN) return;      // whole waves retire together

    const int tm = wave / tilesN, tn = wave % tilesN;
    const int m0 = tm << 5, n0 = tn << 5;

    const int lane = threadIdx.x & 31;
    const int half = lane >> 4;               // 0: lanes 0-15, 1: lanes 16-31
    const int lid  = lane & 15;

    // clamped row/col indices: OOB lanes load duplicate in-bounds data whose
    // products land only in accumulator slots never stored by the epilogue.
    const int r0 = m0 + lid,      r1 = m0 + 16 + lid;
    const int c0 = n0 + lid,      c1 = n0 + 16 + lid;
    const int r0c = r0 < M ? r0 : M - 1;
    const int r1c = r1 < M ? r1 : M - 1;
    const int c0c = c0 < N ? c0 : N - 1;
    const int c1c = c1 < N ? c1 : N - 1;

    // ISA f32 fragment layout: lanes 0-15 hold K=k0,k0+1; lanes 16-31 K=k0+2,k0+3
    const float* pa0 = A + (long)r0c * lda + 2 * half;
    const float* pa1 = A + (long)r1c * lda + 2 * half;
    const float* pb0 = B + (long)(2 * half) * ldb + c0c;
    const float* pb1 = B + (long)(2 * half) * ldb + c1c;

    v8f acc00 = {0.f,0.f,0.f,0.f,0.f,0.f,0.f,0.f};
    v8f acc01 = {0.f,0.f,0.f,0.f,0.f,0.f,0.f,0.f};
    v8f acc10 = {0.f,0.f,0.f,0.f,0.f,0.f,0.f,0.f};
    v8f acc11 = {0.f,0.f,0.f,0.f,0.f,0.f,0.f,0.f};

    const int K4 = K & ~3;
    for (int k0 = 0; k0 < K4; k0 += 4) {
        v2f a0, a1, b0, b1;
        a0.x = pa0[0];   a0.y = pa0[1];
        a1.x = pa1[0];   a1.y = pa1[1];
        b0.x = pb0[0];   b0.y = pb0[ldb];
        b1.x = pb1[0];   b1.y = pb1[ldb];
        pa0 += 4; pa1 += 4;
        pb0 += (long)4 * ldb; pb1 += (long)4 * ldb;
        acc00 = __builtin_amdgcn_wmma_f32_16x16x4_f32(false, a0, false, b0, (short)0, acc00, false, false);
        acc01 = __builtin_amdgcn_wmma_f32_16x16x4_f32(false, a0, false, b1, (short)0, acc01, false, false);
        acc10 = __builtin_amdgcn_wmma_f32_16x16x4_f32(false, a1, false, b0, (short)0, acc10, false, false);
        acc11 = __builtin_amdgcn_wmma_f32_16x16x4_f32(false, a1, false, b1, (short)0, acc11, false, false);
    }
    if (K4 < K) {                              // remainder (K=307 case): mask A only
        const int ka  = K4 + 2 * half;
        const float f0 = (ka     < K) ? 1.f : 0.f;
        const float f1 = (ka + 1 < K) ? 1.f : 0.f;
        const int ka0 = (ka     < K) ? ka     : K - 1;
        const int ka1 = (ka + 1 < K) ? ka + 1 : K - 1;
        v2f a0, a1, b0, b1;
        a0.x = A[(long)r0c * lda + ka0] * f0;  a0.y = A[(long)r0c * lda + ka1] * f1;
        a1.x = A[(long)r1c * lda + ka0] * f0;  a1.y = A[(long)r1c * lda + ka1] * f1;
        b0.x = B[(long)ka0 * ldb + c0c];       b0.y = B[(long)ka1 * ldb + c0c];
        b1.x = B[(long)ka0 * ldb + c1c];       b1.y = B[(long)ka1 * ldb + c1c];
        acc00 = __builtin_amdgcn_wmma_f32_16x16x4_f32(false, a0, false, b0, (short)0, acc00, false, false);
        acc01 = __builtin_amdgcn_wmma_f32_16x16x4_f32(false, a0, false, b1, (short)0, acc01, false, false);
        acc10 = __builtin_amdgcn_wmma_f32_16x16x4_f32(false, a1, false, b0, (short)0, acc10, false, false);
        acc11 = __builtin_amdgcn_wmma_f32_16x16x4_f32(false, a1, false, b1, (short)0, acc11, false, false);
    }

    store_tile(C, ldc, acc00, m0,      n0,      M, N, bias, relu, half, lid);
    store_tile(C, ldc, acc01, m0,      n0 + 16, M, N, bias, relu, half, lid);
    store_tile(C, ldc, acc10, m0 + 16, n0,      M, N, bias, relu, half, lid);
    store_tile(C, ldc, acc11, m0 + 16, n0 + 16, M, N, bias, relu, half, lid);
}

// ---------------------------------------------------------------------------
// Small VALU kernels
// ---------------------------------------------------------------------------
__device__ __forceinline__ float sigmoidf_(float v) {
    return 1.f / (1.f + __expf(-v));
}

// f1[b,t,n] = sum_c x[b,c,n,t] * t_c1[c]
__global__ void k_f1(const float* __restrict__ x, const float* __restrict__ w,
                     float* __restrict__ o) {
    int idx = blockIdx.x * blockDim.x + threadIdx.x;
    if (idx >= Bn * Tn * Nn) return;
    int n = idx % Nn, t = (idx / Nn) % Tn, b = idx / (Nn * Tn);
    const float* xp = x + ((long)b * CIN * Nn + n) * Tn + t;
    float s = 0.f;
    for (int c = 0; c < CIN; ++c) s += xp[(long)c * NT] * w[c];
    o[idx] = s;
}

// f2[b,c,t] = sum_n x[b,c,n,t] * t_c2[n]
__global__ void k_f2(const float* __restrict__ x, const float* __restrict__ w,
                     float* __restrict__ o) {
    int idx = blockIdx.x * blockDim.x + threadIdx.x;
    if (idx >= Bn * CIN * Tn) return;
    int t = idx % Tn, c = (idx / Tn) % CIN, b = idx / (Tn * CIN);
    const float* xp = x + ((long)b * CIN + c) * NT + t;
    float s = 0.f;
    for (int n = 0; n < Nn; ++n) s += xp[(long)n * Tn] * w[n];
    o[idx] = s;
}

// g1[b,n,t] = sum_c x[b,c,n,t] * s_c1[c]
__global__ void k_g1(const float* __restrict__ x, const float* __restrict__ w,
                     float* __restrict__ o) {
    int idx = blockIdx.x * blockDim.x + threadIdx.x;
    if (idx >= Bn * Nn * Tn) return;
    int t = idx % Tn, n = (idx / Tn) % Nn, b = idx / NT;
    const float* xp = x + ((long)b * CIN * Nn + n) * Tn + t;
    float s = 0.f;
    for (int c = 0; c < CIN; ++c) s += xp[(long)c * NT] * w[c];
    o[idx] = s;
}

// g2[b,c,n] = sum_t x[b,c,n,t] * s_c2[t]
__global__ void k_g2(const float* __restrict__ x, const float* __restrict__ w,
                     float* __restrict__ o) {
    int idx = blockIdx.x * blockDim.x + threadIdx.x;
    if (idx >= Bn * CIN * Nn) return;
    int n = idx % Nn, c = (idx / Nn) % CIN, b = idx / (Nn * CIN);
    const float* xp = x + (((long)b * CIN + c) * Nn + n) * Tn;
    float s = 0.f;
    for (int t = 0; t < Tn; ++t) s += xp[t] * w[t];
    o[idx] = s;
}

// tmpT[b,p,c] = sum_n f1[b,p,n] * t_w[n,c]
__global__ void k_tatt_tmp(const float* __restrict__ f1,
                           const float* __restrict__ tw,
                           float* __restrict__ o) {
    int idx = blockIdx.x * blockDim.x + threadIdx.x;
    if (idx >= Bn * Tn * CIN) return;
    int c = idx % CIN, p = (idx / CIN) % Tn, b = idx / (CIN * Tn);
    const float* f = f1 + ((long)b * Tn + p) * Nn;
    float s = 0.f;
    for (int n = 0; n < Nn; ++n) s += f[n] * tw[(long)n * CIN + c];
    o[idx] = s;
}

// lgT[b,p,q] = sigmoid( sum_c tmpT[b,p,c]*f2[b,c,q] + t_b[p,q] )
__global__ void k_tatt_logits(const float* __restrict__ tmpT,
                              const float* __restrict__ f2,
                              const float* __restrict__ tb,
                              float* __restrict__ o) {
    int idx = blockIdx.x * blockDim.x + threadIdx.x;
    if (idx >= Bn * Tn * Tn) return;
    int q = idx % Tn, p = (idx / Tn) % Tn, b = idx / (Tn * Tn);
    const float* tp = tmpT + ((long)b * Tn + p) * CIN;
    const float* fp = f2 + (long)b * CIN * Tn + q;
    float s = tb[p * Tn + q];
    for (int c = 0; c < CIN; ++c) s += tp[c] * fp[(long)c * Tn];
    o[idx] = sigmoidf_(s);
}

// l2T[b,p,q] = sum_n t_v[p,n] * lgT[b,n,q]
__global__ void k_tatt_vmm(const float* __restrict__ tv,
                           const float* __restrict__ lg,
                           float* __restrict__ o) {
    int idx = blockIdx.x * blockDim.x + threadIdx.x;
    if (idx >= Bn * Tn * Tn) return;
    int q = idx % Tn, p = (idx / Tn) % Tn, b = idx / (Tn * Tn);
    const float* lp = lg + (long)b * Tn * Tn + q;
    float s = 0.f;
    for (int n = 0; n < Tn; ++n) s += tv[p * Tn + n] * lp[(long)n * Tn];
    o[idx] = s;
}

// tmpS[b,n,c] = sum_t g1[b,n,t] * s_w[t,c]
__global__ void k_satt_tmp(const float* __restrict__ g1,
                           const float* __restrict__ sw,
                           float* __restrict__ o) {
    int idx = blockIdx.x * blockDim.x + threadIdx.x;
    if (idx >= Bn * Nn * CIN) return;
    int c = idx % CIN, n = (idx / CIN) % Nn, b = idx / (CIN * Nn);
    const float* g = g1 + ((long)b * Nn + n) * Tn;
    float s = 0.f;
    for (int t = 0; t < Tn; ++t) s += g[t] * sw[(long)t * CIN + c];
    o[idx] = s;
}

// in-place: v = sigmoid(v + s_b[p,q])  over [B,N,N]
__global__ void k_sigb(float* __restrict__ v, const float* __restrict__ sb) {
    int idx = blockIdx.x * blockDim.x + threadIdx.x;
    if (idx >= Bn * NNn) return;
    int pq = idx % NNn;
    v[idx] = sigmoidf_(v[idx] + sb[pq]);
}

// column max over p: cm[b,q] = max_p mat[b,p,q]
__global__ void k_colmax(const float* __restrict__ mat, float* __restrict__ cm,
                         int P, int Q) {
    int idx = blockIdx.x * blockDim.x + threadIdx.x;
    if (idx >= Bn * Q) return;
    int q = idx % Q, b = idx / Q;
    const float* m = mat + (long)b * P * Q + q;
    float mx = m[0];
    for (int p = 1; p < P; ++p) mx = fmaxf(mx, m[(long)p * Q]);
    cm[idx] = mx;
}

// row softmax over q of (mat[b,p,q]-cm[b,q]); optional transposed store
__global__ void k_softmax(const float* __restrict__ mat,
                          const float* __restrict__ cm,
                          float* __restrict__ out, int P, int Q, int transposed) {
    int bp = blockIdx.x;
    int b = bp / P, p = bp % P;
    const float* row = mat + ((long)b * P + p) * Q;
    const float* cmb = cm + (long)b * Q;
    __shared__ float red[128];
    int tid = threadIdx.x;
    float mx = -3.0e38f;
    for (int q = tid; q < Q; q += 128) mx = fmaxf(mx, row[q] - cmb[q]);
    red[tid] = mx; __syncthreads();
    for (int s = 64; s > 0; s >>= 1) {
        if (tid < s) red[tid] = fmaxf(red[tid], red[tid + s]);
        __syncthreads();
    }
    mx = red[0]; __syncthreads();
    float sm = 0.f;
    for (int q = tid; q < Q; q += 128) sm += __expf(row[q] - cmb[q] - mx);
    red[tid] = sm; __syncthreads();
    for (int s = 64; s > 0; s >>= 1) {
        if (tid < s) red[tid] += red[tid + s];
        __syncthreads();
    }
    float inv = 1.f / red[0];
    for (int q = tid; q < Q; q += 128) {
        float v = __expf(row[q] - cmb[q] - mx) * inv;
        if (transposed) out[(long)b * P * Q + (long)q * P + p] = v;
        else            out[((long)b * P + p) * Q + q] = v;
    }
}

// L2[q,n] = 2*tmp[q,n] - I
__global__ void k_chebfix(const float* __restrict__ tmp, float* __restrict__ L2) {
    int idx = blockIdx.x * blockDim.x + threadIdx.x;
    if (idx >= NNn) return;
    int n = idx % Nn, q = idx / Nn;
    L2[idx] = 2.f * tmp[idx] - (q == n ? 1.f : 0.f);
}

// Ls[b,k,q,n] = S_coef[b,q,n] * Lk[q,n]   (L0=I, L1=supports, L2=cheb)
__global__ void k_ls(const float* __restrict__ S, const float* __restrict__ sup,
                     const float* __restrict__ L2, float* __restrict__ Ls) {
    long idx = (long)blockIdx.x * blockDim.x + threadIdx.x;
    if (idx >= (long)Bn * Kc * NNn) return;
    int n = (int)(idx % Nn);
    int q = (int)((idx / Nn) % Nn);
    int k = (int)((idx / NNn) % Kc);
    int b = (int)(idx / ((long)Kc * NNn));
    float s = S[(long)b * NNn + (long)q * Nn + n];
    float l = (k == 0) ? (q == n ? 1.f : 0.f)
            : (k == 1) ? sup[(long)q * Nn + n]
                       : L2[(long)q * Nn + n];
    Ls[idx] = s * l;
}

// Xp[b,n,c,t] = x_TAt[b,c,n,t]
__global__ void k_transpose(const float* __restrict__ xi, float* __restrict__ xo) {
    long idx = (long)blockIdx.x * blockDim.x + threadIdx.x;
    if (idx >= (long)Bn * CIN * Nn * Tn) return;
    int t = (int)(idx % Tn);
    int c = (int)((idx / Tn) % CIN);
    int n = (int)((idx / ((long)Tn * CIN)) % Nn);
    int b = (int)(idx / ((long)Tn * CIN * Nn));
    xo[idx] = xi[(((long)b * CIN + c) * Nn + n) * Tn + t];
}

// Wr[o, k*64 + c] = gcn_w[o, c*3 + k]
__global__ void k_wr(const float* __restrict__ gw, float* __restrict__ wr) {
    int idx = blockIdx.x * blockDim.x + threadIdx.x;
    if (idx >= COUT * Kc * CIN) return;
    int r = idx % (Kc * CIN), o = idx / (Kc * CIN);
    int k = r / CIN, c = r % CIN;
    wr[idx] = gw[(long)o * (Kc * CIN) + c * Kc + k];
}

// h[b,o,n,t] = relu( tconv(sg) + tc_b[o] + x_input[b,o,n,t] )
__global__ void k_tconv(const float* __restrict__ sg, const float* __restrict__ tw,
                        const float* __restrict__ tb, const float* __restrict__ xin,
                        float* __restrict__ h) {
    long idx = (long)blockIdx.x * blockDim.x + threadIdx.x;
    if (idx >= (long)Bn * COUT * Nn * Tn) return;
    int t = (int)(idx % Tn);
    int n = (int)((idx / Tn) % Nn);
    int o = (int)((idx / NT) % COUT);
    int b = (int)(idx / ((long)COUT * NT));
    float s = tb[o] + xin[idx];
    for (int o2 = 0; o2 < COUT; ++o2) {
        const float* sp = sg + (((long)b * COUT + o2) * Nn + n) * Tn;
        const float* wp = tw + ((long)o * COUT + o2) * 3;
        float a = wp[1] * sp[t];
        if (t - 1 >= 0) a += wp[0] * sp[t - 1];
        if (t + 1 < Tn) a += wp[2] * sp[t + 1];
        s += a;
    }
    h[idx] = s > 0.f ? s : 0.f;
}

// LayerNorm over [C,N,T] per batch, in place. One block per batch.
__global__ void k_ln(float* __restrict__ h, const float* __restrict__ g,
                     const float* __restrict__ bta) {
    int b = blockIdx.x;
    float* hp = h + (long)b * CNT;
    __shared__ float red[256];
    int tid = threadIdx.x;
    float s = 0.f, s2 = 0.f;
    for (int i = tid; i < CNT; i += 256) { float v = hp[i]; s += v; s2 += v * v; }
    red[tid] = s; __syncthreads();
    for (int st = 128; st > 0; st >>= 1) {
        if (tid < st) red[tid] += red[tid + st];
        __syncthreads();
    }
    float mean = red[0] / (float)CNT; __syncthreads();
    red[tid] = s2; __syncthreads();
    for (int st = 128; st > 0; st >>= 1) {
        if (tid < st) red[tid] += red[tid + st];
        __syncthreads();
    }
    float var = red[0] / (float)CNT - mean * mean;
    float rinv = rsqrtf(var + 1e-5f);
    for (int i = tid; i < CNT; i += 256)
        hp[i] = (hp[i] - mean) * rinv * g[i] + bta[i];
}

// ---------------------------------------------------------------------------
// Host side
// ---------------------------------------------------------------------------
static inline void launch_gemm(hipStream_t s, const float* A, const float* B,
                               float* C, const float* bias,
                               int M, int N, int K, int lda, int ldb, int ldc,
                               long sAo, long sAi, long sBo, long sBi,
                               long sCo, long sCi, int zdiv, int relu, int zc) {
    int tiles = ((M + 31) / 32) * ((N + 31) / 32);   // 32x32 per wave
    dim3 grid((tiles + 7) / 8, 1, zc);
    wmma_gemm_f32<<<grid, 256, 0, s>>>(A, B, C, bias, M, N, K, lda, ldb, ldc,
                                       sAo, sAi, sBo, sBi, sCo, sCi, zdiv, relu);
}

extern "C" void kernel_launch(void* const* d_in, const int* in_sizes, int n_in,
                              void* d_out, int out_size, void* d_ws, size_t ws_size,
                              hipStream_t stream) {
    const float* x       = (const float*)d_in[0];
    const float* supports= (const float*)d_in[1];
    const float* conv1_w = (const float*)d_in[2];
    const float* conv1_b = (const float*)d_in[3];
    const float* t_c1    = (const float*)d_in[4];
    const float* t_c2    = (const float*)d_in[5];
    const float* t_w     = (const float*)d_in[6];
    const float* t_b     = (const float*)d_in[7];
    const float* t_v     = (const float*)d_in[8];
    const float* s_c1    = (const float*)d_in[9];
    const float* s_c2    = (const float*)d_in[10];
    const float* s_w     = (const float*)d_in[11];
    const float* s_b     = (const float*)d_in[12];
    const float* s_v     = (const float*)d_in[13];
    const float* gcn_w   = (const float*)d_in[14];
    const float* gcn_b   = (const float*)d_in[15];
    const float* tc_w    = (const float*)d_in[16];
    const float* tc_b    = (const float*)d_in[17];
    const float* ln_g    = (const float*)d_in[18];
    const float* ln_b    = (const float*)d_in[19];

    float* out1 = (float*)d_out;                       // [B,C,N,T]
    float* out2 = out1 + (long)Bn * COUT * NT;         // S_coef [B,N,N]
    float* out3 = out2 + (long)Bn * NNn;               // T_coef (swapped) [B,T,T]

    // scratch bump allocator (deterministic layout every call)
    float* wsp = (float*)d_ws;
    long off = 0;
    auto alloc = [&](long n) { float* p = wsp + off; off += n; return p; };
    float* x_input = alloc((long)Bn * COUT * NT);
    float* f1    = alloc((long)Bn * Tn * Nn);
    float* f2    = alloc((long)Bn * CIN * Tn);
    float* g1    = alloc((long)Bn * Nn * Tn);
    float* g2    = alloc((long)Bn * CIN * Nn);
    float* tmpT  = alloc((long)Bn * Tn * CIN);
    float* lgT   = alloc((long)Bn * Tn * Tn);
    float* l2T   = alloc((long)Bn * Tn * Tn);
    float* cmT   = alloc((long)Bn * Tn);
    float* tmpS  = alloc((long)Bn * Nn * CIN);
    float* lgS   = alloc((long)Bn * NNn);
    float* l2S   = alloc((long)Bn * NNn);
    float* cmS   = alloc((long)Bn * Nn);
    float* chebt = alloc((long)NNn);
    float* L2c   = alloc((long)NNn);
    float* Ls    = alloc((long)Bn * Kc * NNn);
    float* x_TAt = alloc((long)Bn * CIN * NT);         // later reused as sg
    float* Xp    = alloc((long)Bn * Nn * CIN * Tn);
    float* Y     = alloc((long)Bn * Nn * (Kc * CIN) * Tn);
    float* Wr    = alloc((long)COUT * Kc * CIN);
    float* sg    = x_TAt;                              // alias (x_TAt dead by then)

    const int TPB = 256;
    auto blocks = [&](long n) { return (unsigned)((n + TPB - 1) / TPB); };

    // 1) residual 1x1 conv: x_input[b] = conv1_w[64x64] * x[b][64x7368] + bias
    launch_gemm(stream, conv1_w, x, x_input, conv1_b,
                COUT, NT, CIN, CIN, NT, NT,
                0, 0, (long)CIN * NT, 0, (long)COUT * NT, 0, 1, 0, Bn);

    // 2) attention features
    k_f1<<<blocks((long)Bn * Tn * Nn), TPB, 0, stream>>>(x, t_c1, f1);
    k_f2<<<blocks((long)Bn * CIN * Tn), TPB, 0, stream>>>(x, t_c2, f2);
    k_g1<<<blocks((long)Bn * Nn * Tn), TPB, 0, stream>>>(x, s_c1, g1);
    k_g2<<<blocks((long)Bn * CIN * Nn), TPB, 0, stream>>>(x, s_c2, g2);

    // 3) temporal attention -> out3 (stored already swapped: out3[b,l,q]=att[b,q,l])
    k_tatt_tmp<<<blocks((long)Bn * Tn * CIN), TPB, 0, stream>>>(f1, t_w, tmpT);
    k_tatt_logits<<<blocks((long)Bn * Tn * Tn), TPB, 0, stream>>>(tmpT, f2, t_b, lgT);
    k_tatt_vmm<<<blocks((long)Bn * Tn * Tn), TPB, 0, stream>>>(t_v, lgT, l2T);
    k_colmax<<<blocks((long)Bn * Tn), TPB, 0, stream>>>(l2T, cmT, Tn, Tn);
    k_softmax<<<Bn * Tn, 128, 0, stream>>>(l2T, cmT, out3, Tn, Tn, /*transposed=*/1);

    // 4) spatial attention -> out2
    k_satt_tmp<<<blocks((long)Bn * Nn * CIN), TPB, 0, stream>>>(g1, s_w, tmpS);
    launch_gemm(stream, tmpS, g2, lgS, nullptr,
                Nn, Nn, CIN, CIN, Nn, Nn,
                (long)Nn * CIN, 0, (long)CIN * Nn, 0, (long)NNn, 0, 1, 0, Bn);
    k_sigb<<<blocks((long)Bn * NNn), TPB, 0, stream>>>(lgS, s_b);
    launch_gemm(stream, s_v, lgS, l2S, nullptr,
                Nn, Nn, Nn, Nn, Nn, Nn,
                0, 0, (long)NNn, 0, (long)NNn, 0, 1, 0, Bn);
    k_colmax<<<blocks((long)Bn * Nn), TPB, 0, stream>>>(l2S, cmS, Nn, Nn);
    k_softmax<<<Bn * Nn, 128, 0, stream>>>(l2S, cmS, out2, Nn, Nn, /*transposed=*/0);

    // 5) Chebyshev L2 = 2*supports@supports - I
    launch_gemm(stream, supports, supports, chebt, nullptr,
                Nn, Nn, Nn, Nn, Nn, Nn, 0, 0, 0, 0, 0, 0, 1, 0, 1);
    k_chebfix<<<blocks((long)NNn), TPB, 0, stream>>>(chebt, L2c);

    // 6) scaled Laplacians Ls[b,k,q,n] = S_coef[b,q,n] * Lk[q,n]
    k_ls<<<blocks((long)Bn * Kc * NNn), TPB, 0, stream>>>(out2, supports, L2c, Ls);

    // 7) x_TAt[b,c,n,q] = sum_l x[b,c,n,l] * out3[b,l,q]   ([19648x24]x[24x24])
    launch_gemm(stream, x, out3, x_TAt, nullptr,
                CIN * Nn, Tn, Tn, Tn, Tn, Tn,
                (long)CIN * NT, 0, (long)Tn * Tn, 0, (long)CIN * NT, 0, 1, 0, Bn);

    // 8) transpose to [B,N,C,T] for clean graph-conv GEMM
    k_transpose<<<blocks((long)Bn * CIN * NT), TPB, 0, stream>>>(x_TAt, Xp);

    // 9) big contraction: per (b,k): Y = Ls[b,k] (307x307) @ Xp[b] (307x1536)
    //    Y stored as [b][q][k*64+c][t] via ldc=4608, per-k base offset 1536.
    launch_gemm(stream, Ls, Xp, Y, nullptr,
                Nn, CIN * Tn, Nn,
                Nn, CIN * Tn, Kc * CIN * Tn,
                (long)Kc * NNn, (long)NNn,
                (long)Nn * CIN * Tn, 0,
                (long)Nn * Kc * CIN * Tn, (long)CIN * Tn,
                Kc, 0, Bn * Kc);

    // 10) reorder gcn weights, then GCN GEMM per (b,n): [64x192]x[192x24] +bias +relu
    k_wr<<<blocks((long)COUT * Kc * CIN), TPB, 0, stream>>>(gcn_w, Wr);
    launch_gemm(stream, Wr, Y, sg, gcn_b,
                COUT, Tn, Kc * CIN,
                Kc * CIN, Tn, NT,
                0, 0,
                (long)Nn * Kc * CIN * Tn, (long)Kc * CIN * Tn,
                (long)COUT * NT, (long)Tn,
                Nn, 1, Bn * Nn);

    // 11) temporal conv + residual + relu -> out1
    k_tconv<<<blocks((long)Bn * COUT * NT), TPB, 0, stream>>>(sg, tc_w, tc_b,
                                                              x_input, out1);
    // 12) LayerNorm over [C,N,T] per batch, in place
    k_ln<<<Bn, 256, 0, stream>>>(out1, ln_g, ln_b);
}